// GCN_57449482551753
// MI455X (gfx1250) — compile-verified
//
#include <hip/hip_runtime.h>
#include <math.h>

// CDNA5 / gfx1250 GCN forward.
// Workspace layout (floats):
//   dinv : [0, 1<<17)                      (N=100000 used)
//   h1   : N*128                            (X @ W1)
//   agg1 : N*128                            (aggregated, then relu'd hidden)
//   h2   : N*16                             (hidden @ W2)
// Total ~104.3 MiB.

typedef __attribute__((ext_vector_type(2))) float v2f;
typedef __attribute__((ext_vector_type(8))) float v8f;

__device__ __forceinline__ void atomicAddF32(float* p, float v) {
  // relaxed/agent scope -> should lower to no-return global_atomic_add_f32
  (void)__hip_atomic_fetch_add(p, v, __ATOMIC_RELAXED, __HIP_MEMORY_SCOPE_AGENT);
}

// ---------------- degree / normalization ----------------
__global__ void k_init_deg(float* __restrict__ deg, int n) {
  int i = blockIdx.x * blockDim.x + threadIdx.x;
  if (i < n) deg[i] = 1.0f;  // self loop counts once
}

__global__ void k_count_deg(const long long* __restrict__ dst, float* __restrict__ deg,
                            long long e) {
  long long i = (long long)blockIdx.x * blockDim.x + threadIdx.x;
  if (i < e) atomicAddF32(&deg[(int)dst[i]], 1.0f);
}

__global__ void k_rsqrt_inplace(float* __restrict__ deg, int n) {
  int i = blockIdx.x * blockDim.x + threadIdx.x;
  if (i < n) deg[i] = rsqrtf(deg[i]);  // deg >= 1 always
}

// ---------------- f32 WMMA GEMM: C[M,Ncols] = A[M,K] @ B[K,Ncols] ----------------
// One wave per 16x16 output tile. blockIdx.x = row tile (M must be multiple of 16),
// wave id within block = column tile. K must be a multiple of 4.
// A 16x4 fragment (ISA 7.12.2): lanes 0-15 hold row M=lane, K = k0,k0+1;
// lanes 16-31 hold row M=lane-16, K = k0+2,k0+3. B mirrored with N<->M.
// C/D: VGPR r, lanes 0-15 -> (M=r, N=lane); lanes 16-31 -> (M=r+8, N=lane-16).
__global__ void k_gemm_wmma_f32(const float* __restrict__ A, const float* __restrict__ B,
                                float* __restrict__ C, int K, int Ncols) {
  const int lane = threadIdx.x & 31;
  const int wv   = threadIdx.x >> 5;
  const int m0   = blockIdx.x << 4;
  const int n0   = wv << 4;
  const int half = lane >> 4;   // 0 or 1
  const int l15  = lane & 15;

  const float* arow = A + (size_t)(m0 + l15) * K;

  v8f acc = {};
  for (int k0 = 0; k0 < K; k0 += 4) {
    const int kk = k0 + 2 * half;
    v2f a, b;
    a[0] = arow[kk];
    a[1] = arow[kk + 1];
    b[0] = B[(size_t)kk * Ncols + n0 + l15];
    b[1] = B[(size_t)(kk + 1) * Ncols + n0 + l15];
    acc = __builtin_amdgcn_wmma_f32_16x16x4_f32(false, a, false, b, (short)0, acc,
                                                false, false);
  }

  const int row0 = m0 + half * 8;
#pragma unroll
  for (int r = 0; r < 8; ++r) {
    C[(size_t)(row0 + r) * Ncols + n0 + l15] = acc[r];
  }
}

// ---------------- self-loop init (avoids atomics for the diagonal) ----------------
__global__ void k_selfloop_128(const float* __restrict__ h, const float* __restrict__ dinv,
                               float* __restrict__ agg, int n) {
  long long i = (long long)blockIdx.x * blockDim.x + threadIdx.x;
  if (i < (long long)n * 128) {
    const int node = (int)(i >> 7);
    const float di = dinv[node];
    agg[i] = h[i] * di * di;
  }
}

__global__ void k_selfloop_bias_16(const float* __restrict__ h, const float* __restrict__ dinv,
                                   const float* __restrict__ bias, float* __restrict__ outp,
                                   int n) {
  long long i = (long long)blockIdx.x * blockDim.x + threadIdx.x;
  if (i < (long long)n * 16) {
    const int node = (int)(i >> 4);
    const float di = dinv[node];
    outp[i] = h[i] * di * di + bias[(int)(i & 15)];
  }
}

// ---------------- edge scatter (the roofline-dominant phase, L2-resident) ----------------
// One wave (32 lanes) per edge; each lane owns 4 consecutive features.
__global__ void k_edge_scatter_128(const long long* __restrict__ srcA,
                                   const long long* __restrict__ dstA,
                                   const float* __restrict__ dinv,
                                   const float* __restrict__ h,
                                   float* __restrict__ agg, long long E) {
  const long long e = (long long)blockIdx.x * 8 + (threadIdx.x >> 5);
  if (e >= E) return;  // wave-uniform
  const int lane = threadIdx.x & 31;
  const int s = (int)srcA[e];
  const int d = (int)dstA[e];
  const float w = dinv[s] * dinv[d];
  const float4 v = *(const float4*)(h + (size_t)s * 128 + lane * 4);
  float* o = agg + (size_t)d * 128 + lane * 4;
  atomicAddF32(o + 0, v.x * w);
  atomicAddF32(o + 1, v.y * w);
  atomicAddF32(o + 2, v.z * w);
  atomicAddF32(o + 3, v.w * w);
}

// Half... 16-lane group per edge; each lane owns one of 16 classes.
__global__ void k_edge_scatter_16(const long long* __restrict__ srcA,
                                  const long long* __restrict__ dstA,
                                  const float* __restrict__ dinv,
                                  const float* __restrict__ h2,
                                  float* __restrict__ outp, long long E) {
  const long long e = (long long)blockIdx.x * 16 + (threadIdx.x >> 4);
  if (e >= E) return;
  const int c = threadIdx.x & 15;
  const int s = (int)srcA[e];
  const int d = (int)dstA[e];
  const float w = dinv[s] * dinv[d];
  atomicAddF32(outp + (size_t)d * 16 + c, h2[(size_t)s * 16 + c] * w);
}

// ---------------- bias + relu (layer 1 epilogue, in place) ----------------
__global__ void k_bias_relu_128(float* __restrict__ agg, const float* __restrict__ bias,
                                int n) {
  long long i = (long long)blockIdx.x * blockDim.x + threadIdx.x;
  if (i < (long long)n * 128) {
    const float v = agg[i] + bias[(int)(i & 127)];
    agg[i] = v > 0.0f ? v : 0.0f;
  }
}

// ---------------- log-softmax over 16 classes ----------------
__global__ void k_log_softmax16(const float* __restrict__ logits, float* __restrict__ outp,
                                int n) {
  const int i = blockIdx.x * blockDim.x + threadIdx.x;
  if (i >= n) return;
  const float* row = logits + (size_t)i * 16;
  float m = row[0];
#pragma unroll
  for (int c = 1; c < 16; ++c) m = fmaxf(m, row[c]);
  float s = 0.0f;
#pragma unroll
  for (int c = 0; c < 16; ++c) s += __expf(row[c] - m);
  const float lse = m + __logf(s);
  float* o = outp + (size_t)i * 16;
#pragma unroll
  for (int c = 0; c < 16; ++c) o[c] = row[c] - lse;
}

extern "C" void kernel_launch(void* const* d_in, const int* in_sizes, int n_in,
                              void* d_out, int out_size, void* d_ws, size_t ws_size,
                              hipStream_t stream) {
  const float*     x   = (const float*)d_in[0];
  const long long* ei  = (const long long*)d_in[1];   // int64 [2, E]
  const float*     W1  = (const float*)d_in[2];       // [128,128]
  const float*     b1  = (const float*)d_in[3];       // [128]
  const float*     W2  = (const float*)d_in[4];       // [128,16]
  const float*     b2  = (const float*)d_in[5];       // [16]
  float*           out = (float*)d_out;               // [N*16 logits][N*16 logsoftmax]

  const int       N = in_sizes[0] / 128;   // 100000 (multiple of 16)
  const long long E = in_sizes[1] / 2;     // 1600000
  const long long* srcA = ei;
  const long long* dstA = ei + E;

  float* ws   = (float*)d_ws;
  float* dinv = ws;                          // N floats (also deg scratch)
  float* h1   = ws + (1 << 17);              // N*128
  float* agg1 = h1 + (size_t)N * 128;        // N*128
  float* h2   = agg1 + (size_t)N * 128;      // N*16

  const int mtiles = N / 16;                              // 6250
  const unsigned gN    = (unsigned)((N + 255) / 256);
  const unsigned gE    = (unsigned)((E + 255) / 256);
  const unsigned gN128 = (unsigned)(((long long)N * 128 + 255) / 256);
  const unsigned gN16  = (unsigned)(((long long)N * 16 + 255) / 256);

  // degrees -> dinv
  k_init_deg<<<gN, 256, 0, stream>>>(dinv, N);
  k_count_deg<<<gE, 256, 0, stream>>>(dstA, dinv, E);
  k_rsqrt_inplace<<<gN, 256, 0, stream>>>(dinv, N);

  // layer 1: h1 = X @ W1  (8 column tiles -> 8 waves per block)
  k_gemm_wmma_f32<<<mtiles, 256, 0, stream>>>(x, W1, h1, 128, 128);
  k_selfloop_128<<<gN128, 256, 0, stream>>>(h1, dinv, agg1, N);
  k_edge_scatter_128<<<(unsigned)((E + 7) / 8), 256, 0, stream>>>(srcA, dstA, dinv, h1,
                                                                  agg1, E);
  k_bias_relu_128<<<gN128, 256, 0, stream>>>(agg1, b1, N);

  // layer 2: h2 = relu_hidden @ W2  (one 16-wide column tile -> 1 wave per block)
  k_gemm_wmma_f32<<<mtiles, 32, 0, stream>>>(agg1, W2, h2, 128, 16);
  k_selfloop_bias_16<<<gN16, 256, 0, stream>>>(h2, dinv, b2, out, N);
  k_edge_scatter_16<<<(unsigned)((E + 15) / 16), 256, 0, stream>>>(srcA, dstA, dinv, h2,
                                                                   out, E);

  // outputs: logits already in out[0 : N*16); log_softmax into out[N*16 : 2*N*16)
  k_log_softmax16<<<gN, 256, 0, stream>>>(out, out + (size_t)N * 16, N);
}